// CircleProjectionLayer_7954279432753
// MI455X (gfx1250) — compile-verified
//
#include <hip/hip_runtime.h>

// CircleProjectionLayer: z = center + (x - center) * min(1, R / max(||x-center||, 1e-12))
// RADIUS = 1.0, B = 8,388,608 points, layout [B,3] float32 (AoS, 12 B/point).
//
// Pure streaming kernel: ~302 MB of traffic -> ~13 us floor at 23.3 TB/s HBM.
// Strategy: 4 points per thread = 3 x float4 (b128) loads per operand, so each
// wave covers a contiguous 1536 B span per operand -> every 128 B line fully
// used. Output is written with non-temporal (TH=NT) b128 stores so the
// never-re-read result doesn't evict the (re-read across graph replays,
// ~fits-in-192MB-L2) inputs.

typedef float v4f __attribute__((ext_vector_type(4)));

#define RADIUS_F 1.0f

struct P3 { float x, y, z; };

__device__ __forceinline__ P3 project_point(float xx, float xy, float xz,
                                            float cx, float cy, float cz) {
    float dx = xx - cx;
    float dy = xy - cy;
    float dz = xz - cz;
    float n2 = fmaf(dx, dx, fmaf(dy, dy, dz * dz));
    float n  = sqrtf(n2);
    float s  = fminf(1.0f, RADIUS_F / fmaxf(n, 1e-12f));
    P3 r;
    r.x = fmaf(dx, s, cx);
    r.y = fmaf(dy, s, cy);
    r.z = fmaf(dz, s, cz);
    return r;
}

// Main kernel: one thread handles 4 points (12 floats = 3 float4s per array).
__global__ void __launch_bounds__(256)
circle_proj_vec4(const v4f* __restrict__ x,
                 const v4f* __restrict__ c,
                 v4f* __restrict__ out,
                 int ngroups) {
    int g = blockIdx.x * blockDim.x + threadIdx.x;
    if (g >= ngroups) return;

    const int base = 3 * g;
    v4f xa = x[base + 0];
    v4f xb = x[base + 1];
    v4f xc = x[base + 2];
    v4f ca = c[base + 0];
    v4f cb = c[base + 1];
    v4f cc = c[base + 2];

    // point 0: (xa.x, xa.y, xa.z)
    P3 p0 = project_point(xa[0], xa[1], xa[2], ca[0], ca[1], ca[2]);
    // point 1: (xa.w, xb.x, xb.y)
    P3 p1 = project_point(xa[3], xb[0], xb[1], ca[3], cb[0], cb[1]);
    // point 2: (xb.z, xb.w, xc.x)
    P3 p2 = project_point(xb[2], xb[3], xc[0], cb[2], cb[3], cc[0]);
    // point 3: (xc.y, xc.z, xc.w)
    P3 p3 = project_point(xc[1], xc[2], xc[3], cc[1], cc[2], cc[3]);

    v4f oa, ob, oc;
    oa[0] = p0.x; oa[1] = p0.y; oa[2] = p0.z; oa[3] = p1.x;
    ob[0] = p1.y; ob[1] = p1.z; ob[2] = p2.x; ob[3] = p2.y;
    oc[0] = p2.z; oc[1] = p3.x; oc[2] = p3.y; oc[3] = p3.z;

    // Non-temporal 128-bit stores: output is write-once, never re-read on-GPU.
    __builtin_nontemporal_store(oa, &out[base + 0]);
    __builtin_nontemporal_store(ob, &out[base + 1]);
    __builtin_nontemporal_store(oc, &out[base + 2]);
}

// Scalar tail kernel for point counts not divisible by 4 (unused for B = 8.39M,
// but launched deterministically whenever tail > 0).
__global__ void __launch_bounds__(64)
circle_proj_tail(const float* __restrict__ x,
                 const float* __restrict__ c,
                 float* __restrict__ out,
                 int start_pt, int npts) {
    int p = start_pt + blockIdx.x * blockDim.x + threadIdx.x;
    if (p >= npts) return;
    const int b = 3 * p;
    P3 r = project_point(x[b + 0], x[b + 1], x[b + 2],
                         c[b + 0], c[b + 1], c[b + 2]);
    out[b + 0] = r.x;
    out[b + 1] = r.y;
    out[b + 2] = r.z;
}

extern "C" void kernel_launch(void* const* d_in, const int* in_sizes, int n_in,
                              void* d_out, int out_size, void* d_ws, size_t ws_size,
                              hipStream_t stream) {
    const float* x = (const float*)d_in[0];      // [B,3] f32
    const float* c = (const float*)d_in[1];      // [B,3] f32
    float* out     = (float*)d_out;              // [B,3] f32

    const int nfloats = in_sizes[0];             // B*3
    const int npts    = nfloats / 3;             // B
    const int ngroups = npts / 4;                // groups of 4 points (12 floats)
    const int tail    = npts - 4 * ngroups;

    if (ngroups > 0) {
        const int threads = 256;
        const int blocks  = (ngroups + threads - 1) / threads;
        circle_proj_vec4<<<blocks, threads, 0, stream>>>(
            (const v4f*)x, (const v4f*)c, (v4f*)out, ngroups);
    }
    if (tail > 0) {
        const int threads = 64;
        const int blocks  = (tail + threads - 1) / threads;
        circle_proj_tail<<<blocks, threads, 0, stream>>>(
            x, c, out, 4 * ngroups, npts);
    }
}